// BurgersModelWrapper_1288490188837
// MI455X (gfx1250) — compile-verified
//
#include <hip/hip_runtime.h>
#include <hip/hip_bf16.h>

// Burgers PINN on gfx1250: 9-layer tanh MLP + 2nd-order forward-mode AD,
// all matmuls via V_WMMA_F32_16X16X4_F32 (full fp32, matches reference).
// One 16-point tile per wave iteration; weights live in registers as
// zero-padded 16x12 B-operand tiles; inter-layer D->A relayout via LDS.
// tanh is a branchless v_exp_f32/v_rcp_f32 sequence (no EXEC churn between
// WMMAs, unlike ocml tanhf which branches per element).

typedef __attribute__((ext_vector_type(2))) float v2f;
typedef __attribute__((ext_vector_type(8))) float v8f;

#define NU_CONST 0.003183098861837907f   // 0.01/pi

struct Params {
  const float* W[9];
  const float* B[9];
};

__device__ __forceinline__ v8f zero8() {
  v8f z = {0.f,0.f,0.f,0.f,0.f,0.f,0.f,0.f};
  return z;
}

__device__ __forceinline__ v8f wmma4(v2f a, v2f b, v8f c) {
  // D = A(16x4,f32) * B(4x16,f32) + C(16x16,f32)
  return __builtin_amdgcn_wmma_f32_16x16x4_f32(
      /*neg_a=*/false, a, /*neg_b=*/false, b,
      /*c_mod=*/(short)0, c, /*reuse_a=*/false, /*reuse_b=*/false);
}

// Branchless tanh: tanh(x) = sign(x) * (1 - 2/(exp2(2*log2e*|x|) + 1)).
// Lowers to v_exp_f32 + v_rcp_f32 + a few VALU ops; saturates to +-1.
__device__ __forceinline__ float fast_tanh(float x) {
  float ax = __builtin_fabsf(x);
  float e  = __builtin_amdgcn_exp2f(ax * 2.885390081777927f);  // 2/ln2
  float r  = __builtin_amdgcn_rcpf(e + 1.0f);
  float t  = 1.0f - 2.0f * r;
  return __builtin_copysignf(t, x);
}

#define ROWS 12            // K padded 10 -> 12 (3 chunks of K=4)
#define TBUF (16 * ROWS)   // floats per 16xROWS tile buffer

// Store a C/D-layout tile (vgpr j holds M=j+8*half, N=l15) into LDS as
// [point][feat] rows of ROWS floats. Cols 12..15 of the tile are exactly
// zero by weight zero-padding, so only n<12 is kept (stride-12 rows are
// bank-conflict-free on 64 banks for both the stores and the b64 reloads).
__device__ __forceinline__ void spill_tile(float* __restrict__ buf, v8f t,
                                           int half, int l15) {
  if (l15 < ROWS) {
#pragma unroll
    for (int j = 0; j < 8; ++j)
      buf[(j + 8 * half) * ROWS + l15] = t[j];
  }
}

// Reload as A-operand chunks: lane holds M=l15, K = 4c + 2*half + {0,1}.
__device__ __forceinline__ void load_amat(const float* __restrict__ buf,
                                          v2f a[3], int half, int l15) {
#pragma unroll
  for (int c = 0; c < 3; ++c) {
    int k = 4 * c + 2 * half;           // even -> 8B aligned -> ds_load_b64
    a[c].x = buf[l15 * ROWS + k];
    a[c].y = buf[l15 * ROWS + k + 1];
  }
}

template <bool FULL>
__global__ __launch_bounds__(256) void pinn_mlp_kernel(
    const float* __restrict__ X, int npts, Params P,
    float* __restrict__ u_out, float* __restrict__ f_out,
    float* __restrict__ ux_out, float* __restrict__ uxx_out) {
  const int lane = threadIdx.x & 31;
  const int wv   = threadIdx.x >> 5;
  const int half = lane >> 4;
  const int l15  = lane & 15;

  __shared__ float smem[8 * 4 * TBUF];   // 8 waves x 4 tile buffers
  float* bufH   = &smem[wv * 4 * TBUF];
  float* bufDx  = bufH + TBUF;
  float* bufDt  = bufH + 2 * TBUF;
  float* bufDxx = bufH + 3 * TBUF;

  // ---- preload zero-padded weight tiles (B-operand layout) + biases ----
  const int fi[9] = {2, 10, 10, 10, 10, 10, 10, 10, 10};
  const int fo[9] = {10, 10, 10, 10, 10, 10, 10, 10, 1};
  v2f  w[9][3];
  float bias[9];
#pragma unroll
  for (int l = 0; l < 9; ++l) {
#pragma unroll
    for (int c = 0; c < 3; ++c) {
      v2f t; t.x = 0.f; t.y = 0.f;
      int r0 = 4 * c + 2 * half;         // B rows this lane supplies
      if (l15 < fo[l]) {
        if (r0     < fi[l]) t.x = P.W[l][ r0      * fo[l] + l15];
        if (r0 + 1 < fi[l]) t.y = P.W[l][(r0 + 1) * fo[l] + l15];
      }
      w[l][c] = t;
    }
    bias[l] = (l15 < fo[l]) ? P.B[l][l15] : 0.f;
  }

  const int ntile = npts >> 4;
  const int gw    = (gridDim.x * blockDim.x) >> 5;
  const int wid   = (blockIdx.x * blockDim.x + threadIdx.x) >> 5;

  for (int tile = wid; tile < ntile; tile += gw) {   // wave-uniform loop
    const int base = tile << 4;

    // ---- layer 1: A tiles built in-register (K=2 real rows, rest zero) ----
    v2f aH = {0.f, 0.f}, aDx = {0.f, 0.f}, aDt = {0.f, 0.f};
    if (half == 0) {                      // lanes 0-15 hold K=0,1 of point l15
      aH.x  = X[(base + l15) * 2 + 0];    // x
      aH.y  = X[(base + l15) * 2 + 1];    // t
      aDx.x = 1.f;                        // tangent e_x
      aDt.y = 1.f;                        // tangent e_t
    }
    v8f Z = zero8(), Zx = zero8(), Zt = zero8(), Zxx = zero8();
    Z = wmma4(aH, w[0][0], Z);
    if (FULL) {
      Zx = wmma4(aDx, w[0][0], Zx);
      Zt = wmma4(aDt, w[0][0], Zt);
      // Zxx == 0 after an affine layer with zero 2nd-order seed
    }

    v8f H, Dx, Dt, Dxx;
#pragma unroll
    for (int j = 0; j < 8; ++j) {
      float z = Z[j] + bias[0];
      float h = fast_tanh(z);
      H[j] = h;
      if (FULL) {
        float s  = 1.f - h * h;
        float gx = s * Zx[j];
        Dx[j]  = gx;
        Dt[j]  = s * Zt[j];
        Dxx[j] = -2.f * h * gx * Zx[j];   // s*zxx - 2h*s*zx^2 with zxx=0
      }
    }

    // ---- layers 2..9 ----
#pragma unroll
    for (int l = 1; l < 9; ++l) {
      spill_tile(bufH, H, half, l15);
      if (FULL) {
        spill_tile(bufDx,  Dx,  half, l15);
        spill_tile(bufDt,  Dt,  half, l15);
        spill_tile(bufDxx, Dxx, half, l15);
      }
      asm volatile("s_wait_dscnt 0" ::: "memory");   // CDNA5 split DS counter

      v2f aH2[3], aDx2[3], aDt2[3], aDxx2[3];
      load_amat(bufH, aH2, half, l15);
      if (FULL) {
        load_amat(bufDx,  aDx2,  half, l15);
        load_amat(bufDt,  aDt2,  half, l15);
        load_amat(bufDxx, aDxx2, half, l15);
      }

      Z = zero8(); Zx = zero8(); Zt = zero8(); Zxx = zero8();
#pragma unroll
      for (int c = 0; c < 3; ++c) {
        Z = wmma4(aH2[c], w[l][c], Z);
        if (FULL) {
          Zx  = wmma4(aDx2[c],  w[l][c], Zx);
          Zt  = wmma4(aDt2[c],  w[l][c], Zt);
          Zxx = wmma4(aDxx2[c], w[l][c], Zxx);
        }
      }

      if (l < 8) {                         // hidden layers: tanh + tangents
#pragma unroll
        for (int j = 0; j < 8; ++j) {
          float z = Z[j] + bias[l];
          float h = fast_tanh(z);
          H[j] = h;
          if (FULL) {
            float s  = 1.f - h * h;
            float gx = s * Zx[j];
            Dx[j]  = gx;
            Dt[j]  = s * Zt[j];
            Dxx[j] = s * Zxx[j] - 2.f * h * gx * Zx[j];
          }
        }
      }
    }

    // ---- outputs: column N==0 of the final (linear) layer tiles ----
    if (l15 == 0) {                        // lanes 0 (pts 0-7) and 16 (8-15)
#pragma unroll
      for (int j = 0; j < 8; ++j) {
        int   p = base + j + 8 * half;
        float u = Z[j] + bias[8];
        u_out[p] = u;
        if (FULL) {
          float ux  = Zx[j];
          float ut  = Zt[j];
          float uxx = Zxx[j];
          ux_out[p]  = ux;
          uxx_out[p] = uxx;
          f_out[p]   = ut + u * ux - NU_CONST * uxx;
        }
      }
    }
  }
}

extern "C" void kernel_launch(void* const* d_in, const int* in_sizes, int n_in,
                              void* d_out, int out_size, void* d_ws,
                              size_t ws_size, hipStream_t stream) {
  (void)n_in; (void)out_size; (void)d_ws; (void)ws_size;

  const float* X_f = (const float*)d_in[0];
  const float* x0  = (const float*)d_in[1];
  const float* xbl = (const float*)d_in[2];
  const float* xbr = (const float*)d_in[3];

  Params P;
  for (int i = 0; i < 9; ++i) {
    P.W[i] = (const float*)d_in[4 + 2 * i];
    P.B[i] = (const float*)d_in[5 + 2 * i];
  }

  const int NF = in_sizes[0] / 2;
  const int N0 = in_sizes[1] / 2;
  const int NB = in_sizes[2] / 2;

  float* out   = (float*)d_out;          // concatenated in reference order
  float* u_f   = out;
  float* u_0   = u_f + NF;
  float* u_bl  = u_0 + N0;
  float* u_br  = u_bl + NB;
  float* f_f   = u_br + NB;
  float* ux_f  = f_f + NF;
  float* uxx_f = ux_f + NF;

  dim3 blk(256);                          // 8 wave32 per block

  int blocksF = (NF / 16 + 7) / 8;
  if (blocksF > 2048) blocksF = 2048;
  if (blocksF < 1) blocksF = 1;
  pinn_mlp_kernel<true><<<blocksF, blk, 0, stream>>>(X_f, NF, P, u_f, f_f,
                                                     ux_f, uxx_f);

  int blocks0 = (N0 / 16 + 7) / 8;
  if (blocks0 > 2048) blocks0 = 2048;
  if (blocks0 < 1) blocks0 = 1;
  pinn_mlp_kernel<false><<<blocks0, blk, 0, stream>>>(x0, N0, P, u_0, nullptr,
                                                      nullptr, nullptr);

  int blocksB = (NB / 16 + 7) / 8;
  if (blocksB > 2048) blocksB = 2048;
  if (blocksB < 1) blocksB = 1;
  pinn_mlp_kernel<false><<<blocksB, blk, 0, stream>>>(xbl, NB, P, u_bl, nullptr,
                                                      nullptr, nullptr);
  pinn_mlp_kernel<false><<<blocksB, blk, 0, stream>>>(xbr, NB, P, u_br, nullptr,
                                                      nullptr, nullptr);
}